// QuixerCore_TQ_47493748359736
// MI455X (gfx1250) — compile-verified
//
#include <hip/hip_runtime.h>
#include <math.h>

#define NQ    10
#define DIM   1024        // 2^NQ
#define NT    32
#define BATCH 128
#define DM    512
#define NPQC  40
#define BL    (BATCH * NT)   // 4096

typedef float v2f __attribute__((ext_vector_type(2)));
typedef float v8f __attribute__((ext_vector_type(8)));

// ---------------------------------------------------------------------------
// Gate spec for ansatz14 (q=10, 1 layer), 40 gates:
//  k= 0.. 9 : RY(wire k)
//  k=10..19 : CRX(c=i, t=(i+1)%10), i = 9..0
//  k=20..29 : RY(wire k-20)
//  k=30..39 : CRX(c=i, t=(i-1)%10), i = 9,0,1,...,8
// ---------------------------------------------------------------------------
__constant__ int g_type[NPQC] = {
    0,0,0,0,0,0,0,0,0,0,  1,1,1,1,1,1,1,1,1,1,
    0,0,0,0,0,0,0,0,0,0,  1,1,1,1,1,1,1,1,1,1 };
__constant__ int g_w0[NPQC] = {   // RY wire / CRX control
    0,1,2,3,4,5,6,7,8,9,  9,8,7,6,5,4,3,2,1,0,
    0,1,2,3,4,5,6,7,8,9,  9,0,1,2,3,4,5,6,7,8 };
__constant__ int g_w1[NPQC] = {   // CRX target (dummy for RY)
    0,0,0,0,0,0,0,0,0,0,  0,9,8,7,6,5,4,3,2,1,
    0,0,0,0,0,0,0,0,0,0,  8,9,0,1,2,3,4,5,6,7 };

// ---------------------------------------------------------------------------
// Apply the 40-gate PQC to a state held in LDS (re/im planes).
// Wire w has stride 2^(9-w) in the flattened 1024-amp index.
// ---------------------------------------------------------------------------
__device__ __forceinline__ void apply_pqc(float* re, float* im,
                                          const float* cc, const float* ssn,
                                          int tid) {
  for (int k = 0; k < NPQC; ++k) {
    const float c = cc[k], s = ssn[k];
    const int ty  = g_type[k];
    const int stt = 1 << (9 - (ty ? g_w1[k] : g_w0[k]));   // target stride
    const int stc = ty ? (1 << (9 - g_w0[k])) : 0;         // control stride
    for (int pp = tid; pp < DIM / 2; pp += 256) {
      const int i0 = ((pp & ~(stt - 1)) << 1) | (pp & (stt - 1));
      const int i1 = i0 + stt;
      if (ty == 0) {
        // RY: [[c,-s],[s,c]] (real matrix, acts on re & im alike)
        float r0 = re[i0], r1 = re[i1], m0 = im[i0], m1 = im[i1];
        re[i0] = c * r0 - s * r1;  im[i0] = c * m0 - s * m1;
        re[i1] = s * r0 + c * r1;  im[i1] = s * m0 + c * m1;
      } else if (i0 & stc) {
        // CRX on control=1 half: RX = [[c,-i s],[-i s,c]]
        float r0 = re[i0], m0 = im[i0], r1 = re[i1], m1 = im[i1];
        re[i0] = c * r0 + s * m1;  im[i0] = c * m0 - s * r1;
        re[i1] = c * r1 + s * m0;  im[i1] = c * m1 - s * r0;
      }
    }
    __syncthreads();
  }
}

// ---------------------------------------------------------------------------
// Zero helper (atomics need a clean accumulator).
// ---------------------------------------------------------------------------
__global__ void zero_f32(float* __restrict__ p, int n) {
  int i = blockIdx.x * blockDim.x + threadIdx.x;
  if (i < n) p[i] = 0.f;
}

// ---------------------------------------------------------------------------
// Normalize LCU weights: lcu = (re + i im) / sum(|re + i im|).  One wave.
// ---------------------------------------------------------------------------
__global__ void prep_lcu(const float* __restrict__ lr, const float* __restrict__ li,
                         float* __restrict__ lcu_w) {
  __shared__ float sm[NT];
  __shared__ float inv;
  int t = threadIdx.x;
  if (t < NT) sm[t] = sqrtf(lr[t] * lr[t] + li[t] * li[t]);
  __syncthreads();
  if (t == 0) {
    float s = 0.f;
    for (int i = 0; i < NT; ++i) s += sm[i];
    inv = 1.f / s;
  }
  __syncthreads();
  if (t < NT) { lcu_w[t] = lr[t] * inv; lcu_w[NT + t] = li[t] * inv; }
}

// ---------------------------------------------------------------------------
// pqc = emb(4096x512) @ W.T(512x40) + b using V_WMMA_F32_16X16X4_F32.
// One wave per 16x16 output tile; K loop in steps of 8 (two WMMA chains).
// A lane layout (16x4 f32): lane&15 = M, lane>>4 selects K pair {0,1}/{2,3}.
// B lane layout (4x16 f32): lane&15 = N, same K-pair split.
// C/D layout: lane&15 = N, VGPR v -> M = v + 8*(lane>>4).
// Lanes with n >= 40 clamp their B row index (results discarded at store),
// keeping the inner loop divergence-free (no EXEC save/restore per step).
// ---------------------------------------------------------------------------
__global__ void gemm_wmma(const float* __restrict__ emb, const float* __restrict__ W,
                          const float* __restrict__ bias, float* __restrict__ pqc) {
  const int tm   = blockIdx.x;        // 0..255
  const int tn   = blockIdx.y;        // 0..2
  const int lane = threadIdx.x;       // wave32
  const int half = lane >> 4;
  const int l16  = lane & 15;
  const int m    = tm * 16 + l16;                 // A row for this lane
  const int n    = tn * 16 + l16;                 // B col for this lane
  const int ncl  = (n < NPQC) ? n : (NPQC - 1);   // clamped load row

  const float* arow = emb + (size_t)m * DM + half * 2;
  const float* brow = W   + (size_t)ncl * DM + half * 2;

  v8f c0 = {}; v8f c1 = {};
  for (int k0 = 0; k0 < DM; k0 += 8) {
    v2f a0 = *(const v2f*)(arow + k0);
    v2f a1 = *(const v2f*)(arow + k0 + 4);
    v2f b0 = *(const v2f*)(brow + k0);
    v2f b1 = *(const v2f*)(brow + k0 + 4);
    c0 = __builtin_amdgcn_wmma_f32_16x16x4_f32(false, a0, false, b0,
                                               (short)0, c0, false, false);
    c1 = __builtin_amdgcn_wmma_f32_16x16x4_f32(false, a1, false, b1,
                                               (short)0, c1, false, false);
  }
  if (n < NPQC) {
    float bv = bias[n];
#pragma unroll
    for (int v = 0; v < 8; ++v) {
      int mm = tm * 16 + half * 8 + v;
      pqc[mm * NPQC + n] = c0[v] + c1[v] + bv;
    }
  }
}

// ---------------------------------------------------------------------------
// One workgroup per (b,l): evolve state through PQC(params[b,l]) in LDS,
// then atomically accumulate lcu[l] * state into dst[b].
// init_mode=1: start from |0...0>;  else read prev mono[b] from src (b128).
// ---------------------------------------------------------------------------
__global__ void __launch_bounds__(256)
pqc_lcu(const float* __restrict__ pqc, const float* __restrict__ lcu_w,
        const float* __restrict__ src, float* __restrict__ dst, int init_mode) {
  __shared__ float re[DIM], im[DIM];
  __shared__ float cc[NPQC], ssn[NPQC];
  const int bl  = blockIdx.x;
  const int b   = bl >> 5;
  const int l   = bl & 31;
  const int tid = threadIdx.x;

  const float* prm = pqc + bl * NPQC;
  if (tid < NPQC) {
    float h = 0.5f * prm[tid];
    cc[tid] = cosf(h);  ssn[tid] = sinf(h);
  }
  if (init_mode) {
    for (int d = tid; d < DIM; d += 256) { re[d] = (d == 0) ? 1.f : 0.f; im[d] = 0.f; }
  } else {
    // b128 loads: two complex amps per thread-iteration
    const float4* s4 = (const float4*)src;
    for (int d2 = tid; d2 < DIM / 2; d2 += 256) {
      float4 v = s4[(size_t)b * (DIM / 2) + d2];
      re[2 * d2]     = v.x;  im[2 * d2]     = v.y;
      re[2 * d2 + 1] = v.z;  im[2 * d2 + 1] = v.w;
    }
  }
  __syncthreads();

  apply_pqc(re, im, cc, ssn, tid);

  const float wr = lcu_w[l], wi = lcu_w[NT + l];
  for (int d = tid; d < DIM; d += 256) {
    float r = re[d], m = im[d];
    atomicAdd(dst + (size_t)(b * DIM + d) * 2,     wr * r - wi * m);
    atomicAdd(dst + (size_t)(b * DIM + d) * 2 + 1, wr * m + wi * r);
  }
}

// ---------------------------------------------------------------------------
// acc = q0*e0 + q1*mono1 + q2*mono2 ; normalize (sum|qsvt| cancels against
// the L2 norm); apply shared-parameter PQC; measure <X>,<Y>,<Z> per qubit.
// ---------------------------------------------------------------------------
__global__ void __launch_bounds__(256)
finalize(const float* __restrict__ mono1, const float* __restrict__ mono2,
         const float* __restrict__ qsvt, const float* __restrict__ ffp,
         float* __restrict__ out) {
  __shared__ float re[DIM], im[DIM];
  __shared__ float cc[NPQC], ssn[NPQC];
  __shared__ float red[256];
  __shared__ float rx[256], ry[256], rz[256];
  __shared__ float inv;
  const int b = blockIdx.x, tid = threadIdx.x;

  const float q0 = qsvt[0], q1 = qsvt[1], q2 = qsvt[2];
  if (tid < NPQC) {
    float h = 0.5f * ffp[tid];
    cc[tid] = cosf(h);  ssn[tid] = sinf(h);
  }
  const float4* m1p = (const float4*)mono1;
  const float4* m2p = (const float4*)mono2;
  float loc = 0.f;
  for (int d2 = tid; d2 < DIM / 2; d2 += 256) {
    float4 a = m1p[(size_t)b * (DIM / 2) + d2];
    float4 c2 = m2p[(size_t)b * (DIM / 2) + d2];
    int d = 2 * d2;
    float r0 = q1 * a.x + q2 * c2.x + ((d == 0) ? q0 : 0.f);
    float i0v = q1 * a.y + q2 * c2.y;
    float r1 = q1 * a.z + q2 * c2.z;
    float i1v = q1 * a.w + q2 * c2.w;
    re[d] = r0;     im[d] = i0v;
    re[d + 1] = r1; im[d + 1] = i1v;
    loc += r0 * r0 + i0v * i0v + r1 * r1 + i1v * i1v;
  }
  red[tid] = loc; __syncthreads();
  for (int off = 128; off > 0; off >>= 1) {
    if (tid < off) red[tid] += red[tid + off];
    __syncthreads();
  }
  if (tid == 0) inv = rsqrtf(red[0]);
  __syncthreads();
  const float iv = inv;
  for (int d = tid; d < DIM; d += 256) { re[d] *= iv; im[d] *= iv; }
  __syncthreads();

  apply_pqc(re, im, cc, ssn, tid);

  for (int w = 0; w < NQ; ++w) {
    const int st = 1 << (9 - w);
    float x = 0.f, y = 0.f, z = 0.f;
    for (int pp = tid; pp < DIM / 2; pp += 256) {
      int i0 = ((pp & ~(st - 1)) << 1) | (pp & (st - 1));
      int i1 = i0 + st;
      float r0 = re[i0], m0 = im[i0], r1 = re[i1], m1 = im[i1];
      x += r0 * r1 + m0 * m1;        // Re(conj(a0)*a1)
      y += r0 * m1 - m0 * r1;        // Im(conj(a0)*a1)
      z += r0 * r0 + m0 * m0 - r1 * r1 - m1 * m1;
    }
    rx[tid] = x; ry[tid] = y; rz[tid] = z; __syncthreads();
    for (int off = 128; off > 0; off >>= 1) {
      if (tid < off) {
        rx[tid] += rx[tid + off];
        ry[tid] += ry[tid + off];
        rz[tid] += rz[tid + off];
      }
      __syncthreads();
    }
    if (tid == 0) {
      out[b * 30 + w]          = 2.f * rx[0];
      out[b * 30 + NQ + w]     = 2.f * ry[0];
      out[b * 30 + 2 * NQ + w] = rz[0];
    }
    __syncthreads();
  }
}

// ---------------------------------------------------------------------------
// Host-side launch sequence (graph-capture safe: only kernel launches).
// Workspace layout (floats):
//   [0,163840)          pqc_params (4096 x 40)
//   [163840,163904)     normalized lcu (wr[32], wi[32])
//   [163904,426048)     mono1 (128 x 1024 complex interleaved)
//   [426048,688192)     mono2
// ---------------------------------------------------------------------------
extern "C" void kernel_launch(void* const* d_in, const int* in_sizes, int n_in,
                              void* d_out, int out_size, void* d_ws, size_t ws_size,
                              hipStream_t stream) {
  const float* emb  = (const float*)d_in[0];   // (128, 32, 512)
  const float* W    = (const float*)d_in[1];   // (40, 512)
  const float* bias = (const float*)d_in[2];   // (40,)
  const float* lr   = (const float*)d_in[3];   // (32,)
  const float* li   = (const float*)d_in[4];   // (32,)
  const float* qsvt = (const float*)d_in[5];   // (3,)
  const float* ffp  = (const float*)d_in[6];   // (40,)
  float* out = (float*)d_out;                  // (128, 30)

  float* ws    = (float*)d_ws;
  float* pqc   = ws;
  float* lcu_w = ws + 163840;
  float* mono1 = ws + 163904;
  float* mono2 = ws + 426048;

  // 1. clear accumulators (mono1 + mono2 = 524288 floats, contiguous)
  zero_f32<<<dim3(2048), dim3(256), 0, stream>>>(mono1, 2 * BATCH * DIM * 2);
  // 2. normalize LCU weights
  prep_lcu<<<dim3(1), dim3(32), 0, stream>>>(lr, li, lcu_w);
  // 3. parameter GEMM via f32 WMMA
  gemm_wmma<<<dim3(BL / 16, 3), dim3(32), 0, stream>>>(emb, W, bias, pqc);
  // 4. QSVT monomial k=1: from |0>, accumulate into mono1
  pqc_lcu<<<dim3(BL), dim3(256), 0, stream>>>(pqc, lcu_w, (const float*)nullptr,
                                              mono1, 1);
  // 5. QSVT monomial k=2: from mono1, accumulate into mono2
  pqc_lcu<<<dim3(BL), dim3(256), 0, stream>>>(pqc, lcu_w, mono1, mono2, 0);
  // 6. combine, normalize, final PQC, measure
  finalize<<<dim3(BATCH), dim3(256), 0, stream>>>(mono1, mono2, qsvt, ffp, out);
}